// tf_gather_object_pc_62989990363749
// MI455X (gfx1250) — compile-verified
//
#include <hip/hip_runtime.h>
#include <cstdint>

#define COMPACT_BLOCK 1024     // 32 wave32 waves
#define NUM_POINTS_P  2048

// ---------------------------------------------------------------------------
// Kernel 1: per-batch ordered stream compaction of mask>0.5 indices.
// One block per batch row. Wave32 ballot prefix-sum within waves, small
// serial scan across the 32 wave totals, scatter ranks < P, early exit once
// the running positive count reaches P (exact count beyond P is provably
// unused because j % count == j for j < 2048 <= count).
// ---------------------------------------------------------------------------
__global__ __launch_bounds__(COMPACT_BLOCK)
void compact_mask_kernel(const float* __restrict__ mask,
                         int* __restrict__ idxbuf,
                         unsigned* __restrict__ counts,
                         int N, int P) {
    const int b = blockIdx.x;
    const float* m = mask + (size_t)b * (size_t)N;
    int* ib = idxbuf + (size_t)b * (size_t)P;

    __shared__ unsigned waveSums[COMPACT_BLOCK / 32];
    __shared__ unsigned waveOff [COMPACT_BLOCK / 32];
    __shared__ unsigned runningSh;

    if (threadIdx.x == 0) runningSh = 0u;
    __syncthreads();

    const unsigned lane = threadIdx.x & 31u;       // wave32 (gfx1250)
    const unsigned wid  = threadIdx.x >> 5;
    const unsigned laneMaskLt = (1u << lane) - 1u;

    for (int base = 0; base < N; base += COMPACT_BLOCK) {
        const int n = base + (int)threadIdx.x;
        const bool flag = (n < N) && (m[n] > 0.5f);

        const unsigned bal    = (unsigned)__ballot(flag);  // low 32 bits valid on wave32
        const unsigned prefix = (unsigned)__popc(bal & laneMaskLt);
        if (lane == 0) waveSums[wid] = (unsigned)__popc(bal);
        __syncthreads();

        if (threadIdx.x == 0) {
            unsigned acc = runningSh;
#pragma unroll
            for (int i = 0; i < COMPACT_BLOCK / 32; ++i) {
                waveOff[i] = acc;
                acc += waveSums[i];
            }
            runningSh = acc;
        }
        __syncthreads();

        if (flag) {
            const unsigned rank = waveOff[wid] + prefix;
            if (rank < (unsigned)P) ib[rank] = n;
        }
        __syncthreads();

        if (runningSh >= (unsigned)P) break;  // uniform across block
    }

    if (threadIdx.x == 0) {
        const unsigned c = runningSh;
        counts[b] = (c > (unsigned)P) ? (unsigned)P : c;  // clamp: semantics-preserving
    }
}

// ---------------------------------------------------------------------------
// Kernel 2: gather [B,P,C] from [B,N,C] using compacted indices with
// wrap-around. ONE THREAD PER POINT (64 B). Data path: gfx1250 async DMA
//   4x global_load_async_to_lds_b128 (offset:0/16/32/48)
//   1x s_wait_asynccnt 0
//   4x global_store_async_from_lds_b128 (offset:0/16/32/48)
// The instruction IOFFSET is added to BOTH the LDS and the global address
// (CDNA5 ISA), so one base register pair per
// side covers the whole 64-B point. Tracked by ASYNCcnt; S_ENDPGM implies
// wait-idle so the final stores drain before wave exit.
// ---------------------------------------------------------------------------
__global__ __launch_bounds__(256)
void gather_async_kernel(const float* __restrict__ pc,
                         const int* __restrict__ idxbuf,
                         const unsigned* __restrict__ counts,
                         float* __restrict__ out,
                         int N, int P, int C, int totalPoints) {
    __shared__ float4 bounce[256 * 4];   // 64 B per thread = 16 KB per block

    const int point = blockIdx.x * 256 + (int)threadIdx.x;  // global point id in [0, B*P)
    if (point >= totalPoints) return;

    const int b = point / P;
    const int j = point - b * P;

    const unsigned cnt = counts[b];
    const int col    = (cnt > 0u) ? (j % (int)cnt) : 0;     // wrap-around sampling
    const int srcRow = (cnt > 0u) ? idxbuf[b * P + col] : 0;

    const float* src = pc  + ((size_t)b * (size_t)N + (size_t)srcRow) * (size_t)C;
    float*       dst = out + (size_t)point * (size_t)C;

    // Low 32 bits of a generic pointer into the LDS aperture == workgroup-
    // relative LDS byte address (isLDS aperture: addr[31:0] is the LDS offset).
    const unsigned ldsOff = (unsigned)(uintptr_t)(const void*)&bounce[threadIdx.x * 4];
    const uint64_t srcA = (uint64_t)(uintptr_t)src;
    const uint64_t dstA = (uint64_t)(uintptr_t)dst;

    // 64 B per lane: global -> LDS (async, ASYNCcnt += 4)
    asm volatile(
        "global_load_async_to_lds_b128 %0, %1, off\n\t"
        "global_load_async_to_lds_b128 %0, %1, off offset:16\n\t"
        "global_load_async_to_lds_b128 %0, %1, off offset:32\n\t"
        "global_load_async_to_lds_b128 %0, %1, off offset:48"
        :: "v"(ldsOff), "v"(srcA) : "memory");
    // one wait per point: ASYNCcnt -> 0 (loads landed in LDS)
    asm volatile("s_wait_asynccnt 0x0" ::: "memory");
    // 64 B per lane: LDS -> global (async)
    asm volatile(
        "global_store_async_from_lds_b128 %0, %1, off\n\t"
        "global_store_async_from_lds_b128 %0, %1, off offset:16\n\t"
        "global_store_async_from_lds_b128 %0, %1, off offset:32\n\t"
        "global_store_async_from_lds_b128 %0, %1, off offset:48"
        :: "v"(dstA), "v"(ldsOff) : "memory");
}

// ---------------------------------------------------------------------------
// Host-side launcher
// ---------------------------------------------------------------------------
extern "C" void kernel_launch(void* const* d_in, const int* in_sizes, int n_in,
                              void* d_out, int out_size, void* d_ws, size_t ws_size,
                              hipStream_t stream) {
    (void)n_in; (void)ws_size;

    const float* pc   = (const float*)d_in[0];   // [B, N, C] f32
    const float* mask = (const float*)d_in[1];   // [B, N]    f32
    float*       out  = (float*)d_out;           // [B, P, C] f32

    const int P   = NUM_POINTS_P;
    const long long BN  = (long long)in_sizes[1];
    const long long BNC = (long long)in_sizes[0];
    const int C = (int)(BNC / BN);               // 16
    const int B = out_size / (P * C);            // 32
    const int N = (int)(BN / (long long)B);      // 262144

    // Scratch layout: [0,128) counts (B u32), [256, 256 + B*P*4) index buffer
    unsigned* counts = (unsigned*)d_ws;
    int*      idxbuf = (int*)((char*)d_ws + 256);

    compact_mask_kernel<<<B, COMPACT_BLOCK, 0, stream>>>(mask, idxbuf, counts, N, P);

    const int totalPoints = B * P;               // 65536 threads, one per 64-B point
    const int blocks = (totalPoints + 255) / 256;
    gather_async_kernel<<<blocks, 256, 0, stream>>>(pc, idxbuf, counts, out, N, P, C, totalPoints);
}